// ContactLoss_87076166959604
// MI455X (gfx1250) — compile-verified
//
#include <hip/hip_runtime.h>
#include <hip/hip_bf16.h>
#include <math.h>

// ---------------------------------------------------------------------------
// ContactLoss for MI455X (gfx1250): nearest-neighbor via V_WMMA_F32_16X16X4_F32
//   dist(m,n) = rx_m + ry_n - 2 h_m . o_n
// WMMA computes (ry - 2 h.o) in one fused 16x16x4 matmul by packing
// A = [-2hx,-2hy,-2hz, 1] and B = [ox; oy; oz; |o|^2]; rx is added post-argmin.
// Object dim split into NSPLIT chunks for occupancy (latency hiding via waves).
// ---------------------------------------------------------------------------

typedef __attribute__((ext_vector_type(2))) float v2f;
typedef __attribute__((ext_vector_type(8))) float v8f;

#define BATCH   8
#define NH      778
#define NO      50000
#define MBLK    ((NH + 15) / 16)      // 49 hand-vertex blocks per batch
#define NTILES  (NO / 16)             // 3125 object tiles (exact)
#define NSPLIT  4                     // object-dim chunks (grid = B*MBLK*NSPLIT)
#define CONTACT_T2  (0.025f * 0.025f)
#define COLL_T      0.025f

// output layout (floats)
#define OFF_CLOSE   2
#define OFF_MINHO   (2 + BATCH * NH * 3)          // 18674
#define OFF_MAXD    (OFF_MINHO + BATCH * NH)      // 24898
#define OFF_MEAND   (OFF_MAXD + 1)                // 24899

// ---------------------------------------------------------------------------
// Kernel 0: pack object points into [x, y, z, |o|^2] float4 (aligned b64 loads
// per half-tile later, and ry precomputed once instead of per hand-block).
// ---------------------------------------------------------------------------
__global__ __launch_bounds__(256) void cl_pack_obj(const float* __restrict__ obj,
                                                   float4* __restrict__ objp) {
    int i = blockIdx.x * 256 + threadIdx.x;
    if (i < BATCH * NO) {
        float x = obj[(size_t)i * 3 + 0];
        float y = obj[(size_t)i * 3 + 1];
        float z = obj[(size_t)i * 3 + 2];
        objp[i] = make_float4(x, y, z, x * x + y * y + z * z);
    }
}

// ---------------------------------------------------------------------------
// Kernel 1: per (16-hand-vertex block, object chunk): WMMA over the chunk's
// tiles with running min/argmin, LDS reduction, write per-chunk candidate.
// Block: 256 threads = 8 waves; wave w of chunk c handles tiles c*8+w + 32*k.
// Prefetches are unconditional: over-reads (<=16KB past a batch's objp) land
// in the following ws regions, which are allocated; values never consumed.
// ---------------------------------------------------------------------------
__global__ __launch_bounds__(256) void cl_nn_wmma(const float* __restrict__ hand,
                                                  const float* __restrict__ objp,
                                                  float* __restrict__ cand_val,
                                                  int*   __restrict__ cand_idx) {
    const int blk   = blockIdx.x;               // ((b*MBLK)+mblk)*NSPLIT + chunk
    const int chunk = blk % NSPLIT;
    const int bm    = blk / NSPLIT;
    const int b     = bm / MBLK;
    const int mBase = (bm % MBLK) * 16;
    const int tid   = threadIdx.x;
    const int lane  = tid & 31;
    // scalar (SGPR) wave index -> scalar loop counter, EXEC untouched in loop
    const int wave  = __builtin_amdgcn_readfirstlane(tid >> 5);
    const int hi    = lane >> 4;          // half-wave: 0 => K rows 0..1, 1 => K rows 2..3
    const int col   = lane & 15;

    // ---- A matrix (16x4): lane (hi,col) holds A[col][2*hi + {0,1}] ----
    float a0 = 0.0f, a1 = (hi ? 1.0f : 0.0f);   // pad row K=3 with 1.0 (multiplies ry)
    {
        int hm = mBase + col;
        if (hm < NH) {
            const float* hp = hand + ((size_t)b * NH + hm) * 3;
            if (hi == 0) { a0 = -2.0f * hp[0]; a1 = -2.0f * hp[1]; }
            else         { a0 = -2.0f * hp[2]; a1 = 1.0f; }
        }
    }
    v2f Av; Av[0] = a0; Av[1] = a1;

    float minv[8];
    int   idxv[8];
#pragma unroll
    for (int v = 0; v < 8; ++v) { minv[v] = 3.0e38f; idxv[v] = 0; }

    // B tile source: packed float4 per point -> two float2 halves per lane
    const float2* bp = (const float2*)(objp + (size_t)b * NO * 4);

    // depth-2 software pipeline, unconditional prefetch (stride 32 tiles)
    int t = chunk * 8 + wave;
    float2 bq0 = bp[(size_t)((t +  0) * 16 + col) * 2 + hi];
    float2 bq1 = bp[(size_t)((t + 32) * 16 + col) * 2 + hi];

    for (; t < NTILES; t += 32) {
        float2 bv = bq0;
        bq0 = bq1;
        bq1 = bp[(size_t)((t + 64) * 16 + col) * 2 + hi];   // may over-read; unused

        v2f Bv; Bv[0] = bv.x; Bv[1] = bv.y;
        v8f c = {0.f, 0.f, 0.f, 0.f, 0.f, 0.f, 0.f, 0.f};
        // D[m][n] = ry_n - 2 h_m . o_n   (EXEC is all-1s here)
        v8f d = __builtin_amdgcn_wmma_f32_16x16x4_f32(
            false, Av, false, Bv, (short)0, c, false, false);

        int n = t * 16 + col;
#pragma unroll
        for (int v = 0; v < 8; ++v) {
            float dv = d[v];
            bool cc = dv < minv[v];           // strict < keeps smallest n in stripe
            idxv[v] = cc ? n : idxv[v];
            minv[v] = cc ? dv : minv[v];
        }
    }

    // ---- cross-lane / cross-wave argmin reduction via LDS ----
    __shared__ float s_val[16][128];
    __shared__ int   s_idx[16][128];
#pragma unroll
    for (int v = 0; v < 8; ++v) {
        int m = v + 8 * hi;                   // C/D layout: M = vgpr + 8*(lane>=16)
        s_val[m][wave * 16 + col] = minv[v];
        s_idx[m][wave * 16 + col] = idxv[v];
    }
    __syncthreads();

    if (tid < 16) {
        int m = tid;
        float best = s_val[m][0];
        int   bi   = s_idx[m][0];
        for (int j = 1; j < 128; ++j) {
            float vv = s_val[m][j];
            int   ii = s_idx[m][j];
            if (vv < best || (vv == best && ii < bi)) { best = vv; bi = ii; }
        }
        int hm = mBase + m;
        if (hm < NH) {
            size_t ci = (size_t)chunk * (BATCH * NH) + (size_t)b * NH + hm;
            cand_val[ci] = best;
            cand_idx[ci] = bi;
        }
    }
}

// ---------------------------------------------------------------------------
// Kernel 1b: merge NSPLIT candidates per hand vertex, gather closest point,
// write results_close / minho / anchor.
// ---------------------------------------------------------------------------
__global__ __launch_bounds__(256) void cl_merge(const float* __restrict__ hand,
                                                const float* __restrict__ objp,
                                                const float* __restrict__ cand_val,
                                                const int*   __restrict__ cand_idx,
                                                float* __restrict__ out_close,
                                                float* __restrict__ out_minho,
                                                float* __restrict__ anchor) {
    int i = blockIdx.x * 256 + threadIdx.x;     // b*NH + m
    if (i >= BATCH * NH) return;
    int b = i / NH;

    float best = cand_val[i];
    int   bi   = cand_idx[i];
#pragma unroll
    for (int ck = 1; ck < NSPLIT; ++ck) {
        float vv = cand_val[(size_t)ck * (BATCH * NH) + i];
        int   ii = cand_idx[(size_t)ck * (BATCH * NH) + i];
        if (vv < best || (vv == best && ii < bi)) { best = vv; bi = ii; }
    }

    const float* hp = hand + (size_t)i * 3;
    float hx = hp[0], hy = hp[1], hz = hp[2];
    float rx = hx * hx + hy * hy + hz * hz;
    const float* op = objp + ((size_t)b * NO + bi) * 4;
    float ox = op[0], oy = op[1], oz = op[2];
    float dx = ox - hx, dy = oy - hy, dz = oz - hz;
    float c2 = dx * dx + dy * dy + dz * dz;
    out_close[(size_t)i * 3 + 0] = ox;
    out_close[(size_t)i * 3 + 1] = oy;
    out_close[(size_t)i * 3 + 2] = oz;
    out_minho[i] = best + rx;                   // minho from the rx+ry-2zz formulation
    anchor[i]    = sqrtf(c2);                   // anchor_dists; contact_vals = anchor^2
}

// ---------------------------------------------------------------------------
// Kernel 2: per-batch masked reductions. One block per batch.
//   partials[b*6 + {0..5}] = {sumMissed, cntMissed, sumPen, cntPen, sumAd, maxAd}
// ---------------------------------------------------------------------------
__global__ __launch_bounds__(256) void cl_batch_reduce(const float* __restrict__ minho,
                                                       const float* __restrict__ anchor,
                                                       const int* __restrict__ exterior,
                                                       float* __restrict__ partials) {
    int b = blockIdx.x;
    int tid = threadIdx.x;
    float sM = 0.f, cM = 0.f, sP = 0.f, cP = 0.f, sA = 0.f, mA = 0.f;
    for (int m = tid; m < NH; m += 256) {
        size_t i = (size_t)b * NH + m;
        float a = anchor[i];
        float d = minho[i];
        int ext = exterior[i];
        bool pm = (ext == 0);                 // penetr_mask = ~exterior
        if ((d < CONTACT_T2) && (ext != 0)) { sM += a * a; cM += 1.0f; }
        float coll = COLL_T * tanhf(a * (1.0f / COLL_T));
        if (pm) { sP += coll; cP += 1.0f; }
        float ad = pm ? a : 0.0f;
        sA += ad;
        mA = fmaxf(mA, ad);
    }
    __shared__ float sh[6][256];
    sh[0][tid] = sM; sh[1][tid] = cM; sh[2][tid] = sP;
    sh[3][tid] = cP; sh[4][tid] = sA; sh[5][tid] = mA;
    __syncthreads();
    for (int s = 128; s > 0; s >>= 1) {
        if (tid < s) {
            sh[0][tid] += sh[0][tid + s];
            sh[1][tid] += sh[1][tid + s];
            sh[2][tid] += sh[2][tid + s];
            sh[3][tid] += sh[3][tid + s];
            sh[4][tid] += sh[4][tid + s];
            sh[5][tid] = fmaxf(sh[5][tid], sh[5][tid + s]);
        }
        __syncthreads();
    }
    if (tid == 0) {
#pragma unroll
        for (int k = 0; k < 6; ++k) partials[b * 6 + k] = sh[k][0];
    }
}

// ---------------------------------------------------------------------------
// Kernel 3: finalize the four scalars.
// ---------------------------------------------------------------------------
__global__ void cl_finalize(const float* __restrict__ partials, float* __restrict__ out) {
    if (threadIdx.x == 0) {
        float sM = 0.f, cM = 0.f, sP = 0.f, cP = 0.f, sA = 0.f, sMax = 0.f;
        for (int b = 0; b < BATCH; ++b) {
            sM += partials[b * 6 + 0];
            cM += partials[b * 6 + 1];
            sP += partials[b * 6 + 2];
            cP += partials[b * 6 + 3];
            sA += partials[b * 6 + 4];
            sMax += partials[b * 6 + 5];
        }
        out[0] = (cM > 0.f) ? (sM / fmaxf(cM, 1.0f)) : 0.0f;   // missed_loss
        out[1] = (cP > 0.f) ? (sP / fmaxf(cP, 1.0f)) : 0.0f;   // penetr_loss
        out[OFF_MAXD]  = sMax / (float)BATCH;                  // mean over b of per-b max
        out[OFF_MEAND] = sA / (float)(BATCH * NH);             // mean over all
    }
}

// ---------------------------------------------------------------------------
extern "C" void kernel_launch(void* const* d_in, const int* in_sizes, int n_in,
                              void* d_out, int out_size, void* d_ws, size_t ws_size,
                              hipStream_t stream) {
    const float* hand = (const float*)d_in[0];   // [B, NH, 3]
    const float* obj  = (const float*)d_in[1];   // [B, NO, 3]
    const int*   ext  = (const int*)d_in[2];     // [B, NH]
    float* out = (float*)d_out;

    // workspace layout (floats):
    //   objp [B*NO*4] | anchor [B*NH] | partials [64] | cand_val [NSPLIT*B*NH]
    //   | cand_idx [NSPLIT*B*NH]
    // (regions after objp also double as harmless over-read padding for the
    //  unconditional prefetch past the last batch's objp region)
    float* ws       = (float*)d_ws;
    float* objp     = ws;
    float* anchor   = ws + (size_t)BATCH * NO * 4;
    float* partials = anchor + (size_t)BATCH * NH;
    float* cand_val = partials + 64;
    int*   cand_idx = (int*)(cand_val + (size_t)NSPLIT * BATCH * NH);

    // 0) pack object points with precomputed |o|^2
    {
        int total = BATCH * NO;
        cl_pack_obj<<<(total + 255) / 256, 256, 0, stream>>>(obj, (float4*)objp);
    }
    // 1) WMMA nearest-neighbor, per-chunk candidates
    cl_nn_wmma<<<BATCH * MBLK * NSPLIT, 256, 0, stream>>>(hand, objp, cand_val, cand_idx);
    // 1b) merge candidates + per-vertex outputs
    {
        int total = BATCH * NH;
        cl_merge<<<(total + 255) / 256, 256, 0, stream>>>(hand, objp, cand_val, cand_idx,
                                                          out + OFF_CLOSE,
                                                          out + OFF_MINHO,
                                                          anchor);
    }
    // 2) per-batch masked reductions
    cl_batch_reduce<<<BATCH, 256, 0, stream>>>(out + OFF_MINHO, anchor, ext, partials);
    // 3) scalars
    cl_finalize<<<1, 32, 0, stream>>>(partials, out);
}